// OhemCrossEntropy2d_42417097016232
// MI455X (gfx1250) — compile-verified
//
#include <hip/hip_runtime.h>
#include <math.h>

// Problem constants from the reference: predict (4,19,512,1024) f32 NCHW,
// target (4,512,1024) int, class_weight (19) f32.
#define NIMG    4
#define NCLASS  19
#define HW      (512 * 1024)          // 524288 = 2^19
#define PIX     (NIMG * HW)           // 2097152 = 2^21
#define IGNORE  255
#define MIN_KEPT 100000
#define THRESH  0.6f

struct Ctrl {
    unsigned int hist[256];
    unsigned int num_valid;
    unsigned int prefix;        // radix-select prefix (grows 8 bits/pass)
    unsigned int kth;           // remaining rank within current prefix
    unsigned int thr_bits;      // float bits of final threshold
    unsigned int apply_filter;  // (num_valid>0) && (num_valid>MIN_KEPT)
    float sum_lw;               // sum(nll * w * kept)
    float sum_w;                // sum(w * kept)
};

typedef __attribute__((ext_vector_type(2))) float v2f;
typedef __attribute__((ext_vector_type(8))) float v8f;

// ---------------------------------------------------------------------------
// Wave32 sum-reduction via V_WMMA_F32_16X16X4_F32.
// A(16x4): VGPR0 lanes0-15 -> (M=lane,K=0), lanes16-31 -> (M=lane-16,K=2);
// VGPR1 -> K=1/K=3 (zeroed). B = ones. D[m][n] = v[m] + v[m+16] for all n.
// Lane L<16 holds D rows 0..7 in c[0..7]; lane L>=16 holds rows 8..15.
// Sum the 8 components, then xor-16 shuffle -> full 32-lane sum in every lane.
// Requires EXEC all-ones (callers use power-of-2 grid-stride loops).
// ---------------------------------------------------------------------------
__device__ __forceinline__ float wave_sum_wmma(float v) {
    v2f a; a[0] = v;    a[1] = 0.0f;
    v2f b; b[0] = 1.0f; b[1] = 1.0f;
    v8f c = {};
    c = __builtin_amdgcn_wmma_f32_16x16x4_f32(
        /*neg_a=*/false, a, /*neg_b=*/false, b,
        /*c_mod=*/(short)0, c, /*reuse_a=*/false, /*reuse_b=*/false);
    float s = c[0] + c[1] + c[2] + c[3] + c[4] + c[5] + c[6] + c[7];
    s += __shfl_xor(s, 16, 32);
    return s;
}

__device__ __forceinline__ void block_accum2(float a, float b,
                                             float* outA, float* outB) {
    __shared__ float sA[8], sB[8];
    float wa = wave_sum_wmma(a);
    float wb = wave_sum_wmma(b);
    int lane = threadIdx.x & 31;
    int wid  = threadIdx.x >> 5;     // blockDim.x == 256 -> 8 waves
    if (lane == 0) { sA[wid] = wa; sB[wid] = wb; }
    __syncthreads();
    if (threadIdx.x == 0) {
        float ta = 0.f, tb = 0.f;
        #pragma unroll
        for (int i = 0; i < 8; ++i) { ta += sA[i]; tb += sB[i]; }
        atomicAdd(outA, ta);
        atomicAdd(outB, tb);
    }
}

// ---------------------------------------------------------------------------
// Kernel 0: re-initialize control block every call (graph replays, ws is
// poisoned once and never re-poisoned).
// ---------------------------------------------------------------------------
__global__ void ohem_init_kernel(Ctrl* __restrict__ ctrl) {
    int t = threadIdx.x;
    ctrl->hist[t] = 0u;
    if (t == 0) {
        ctrl->num_valid = 0u;
        ctrl->prefix = 0u;
        ctrl->kth = 0u;
        ctrl->thr_bits = 0u;
        ctrl->apply_filter = 0u;
        ctrl->sum_lw = 0.f;
        ctrl->sum_w  = 0.f;
    }
}

// ---------------------------------------------------------------------------
// Kernel 1: single pass over predict (159 MB, the HBM-bound part).
// Per pixel: stable softmax over 19 channel-strided logits, store GT-prob
// bits (monotone uint order; invalid -> 0xFFFFFFFF) and NLL; count valid.
// ---------------------------------------------------------------------------
__global__ void ohem_pass1_kernel(const float* __restrict__ predict,
                                  const int*   __restrict__ target,
                                  unsigned int* __restrict__ pred_bits,
                                  float* __restrict__ nll,
                                  Ctrl* __restrict__ ctrl) {
    const int stride = gridDim.x * blockDim.x;
    int cnt = 0;
    for (int p = blockIdx.x * blockDim.x + threadIdx.x; p < PIX; p += stride) {
        int lab = target[p];
        bool valid = (lab != IGNORE);
        int safe = valid ? lab : 0;
        int n  = p >> 19;           // p / HW
        int hw = p & (HW - 1);
        const float* base = predict + (size_t)n * NCLASS * HW + hw;

        float x[NCLASS];
        float m = -3.4e38f;
        float xs = 0.f;
        #pragma unroll
        for (int c = 0; c < NCLASS; ++c) {
            float v = base[(size_t)c * HW];   // 128B/wave coalesced stream
            x[c] = v;
            m = fmaxf(m, v);
            xs = (c == safe) ? v : xs;        // avoid dynamic register index
        }
        float s = 0.f;
        #pragma unroll
        for (int c = 0; c < NCLASS; ++c) s += __expf(x[c] - m);

        float logZ = m + __logf(s);
        nll[p] = logZ - xs;                       // -log_softmax at GT class
        float pr = __expf(xs - m) / s;            // softmax prob at GT class
        pred_bits[p] = valid ? __float_as_uint(pr) : 0xFFFFFFFFu;
        cnt += valid ? 1 : 0;
    }
    // wave32 reduce of the valid count, one atomic per wave
    #pragma unroll
    for (int off = 16; off > 0; off >>= 1) cnt += __shfl_xor(cnt, off, 32);
    if ((threadIdx.x & 31) == 0)
        atomicAdd(&ctrl->num_valid, (unsigned int)cnt);
}

// ---------------------------------------------------------------------------
// Kernel 2 (x4): MSB-first radix-select histogram over the 8 MB key array
// (L2-resident on a 192 MB L2). LDS histogram, then one global atomic/bin.
// ---------------------------------------------------------------------------
__global__ void ohem_hist_kernel(const unsigned int* __restrict__ keys,
                                 Ctrl* __restrict__ ctrl, int pass) {
    __shared__ unsigned int h[256];
    h[threadIdx.x] = 0u;
    __syncthreads();
    const unsigned int prefix = ctrl->prefix;
    const int shift = 24 - 8 * pass;
    const int stride = gridDim.x * blockDim.x;
    for (int p = blockIdx.x * blockDim.x + threadIdx.x; p < PIX; p += stride) {
        unsigned int k = keys[p];
        bool match = (pass == 0) || ((k >> (shift + 8)) == prefix);
        if (match) atomicAdd(&h[(k >> shift) & 0xFFu], 1u);
    }
    __syncthreads();
    unsigned int v = h[threadIdx.x];
    if (v) atomicAdd(&ctrl->hist[threadIdx.x], v);
}

// ---------------------------------------------------------------------------
// Kernel 3 (x4): pick the digit containing rank k; update prefix/k; reset
// histogram for the next pass. On pass 0 derive k and apply_filter from
// num_valid; on pass 3 finalize threshold = max(kth value, 0.6).
// ---------------------------------------------------------------------------
__global__ void ohem_select_kernel(Ctrl* __restrict__ ctrl, int pass) {
    __shared__ unsigned int h[256];
    int t = threadIdx.x;
    h[t] = ctrl->hist[t];
    ctrl->hist[t] = 0u;               // reset for next pass
    __syncthreads();
    if (t == 0) {
        if (pass == 0) {
            unsigned int nv = ctrl->num_valid;
            unsigned int kk = nv < (unsigned int)MIN_KEPT ? nv : (unsigned int)MIN_KEPT;
            ctrl->kth = (kk > 0u) ? (kk - 1u) : 0u;
            ctrl->apply_filter = (nv > 0u && nv > (unsigned int)MIN_KEPT) ? 1u : 0u;
            ctrl->prefix = 0u;
        }
        unsigned int k = ctrl->kth;
        unsigned int cum = 0u;
        int b = 255;
        for (int i = 0; i < 256; ++i) {
            unsigned int nc = cum + h[i];
            if (k < nc) { b = i; break; }
            cum = nc;
        }
        ctrl->prefix = (ctrl->prefix << 8) | (unsigned int)b;
        ctrl->kth = k - cum;
        if (pass == 3) {
            float thr = __uint_as_float(ctrl->prefix);   // exact kth value
            thr = (thr > THRESH) ? thr : THRESH;
            ctrl->thr_bits = __float_as_uint(thr);
        }
    }
}

// ---------------------------------------------------------------------------
// Kernel 4: weighted masked reduction over cached nll/pred (24 MB, mostly L2)
// with the WMMA wave reduction for block accumulation.
// ---------------------------------------------------------------------------
__global__ void ohem_final_kernel(const int*   __restrict__ target,
                                  const float* __restrict__ weight,
                                  const unsigned int* __restrict__ pred_bits,
                                  const float* __restrict__ nll,
                                  Ctrl* __restrict__ ctrl) {
    __shared__ float wtab[NCLASS];
    if (threadIdx.x < NCLASS) wtab[threadIdx.x] = weight[threadIdx.x];
    __syncthreads();
    const float thr = __uint_as_float(ctrl->thr_bits);
    const bool filt = ctrl->apply_filter != 0u;
    float lw = 0.f, wsum = 0.f;
    const int stride = gridDim.x * blockDim.x;   // power of 2: no EXEC ragged edge
    for (int p = blockIdx.x * blockDim.x + threadIdx.x; p < PIX; p += stride) {
        int lab = target[p];
        bool valid = (lab != IGNORE);
        bool kept = valid;
        if (filt && valid) {
            float pr = __uint_as_float(pred_bits[p]);
            kept = (pr <= thr);
        }
        if (kept) {
            float w = wtab[lab];
            wsum += w;
            lw += nll[p] * w;
        }
    }
    block_accum2(lw, wsum, &ctrl->sum_lw, &ctrl->sum_w);
}

__global__ void ohem_finalize_kernel(const Ctrl* __restrict__ ctrl,
                                     float* __restrict__ out) {
    out[0] = ctrl->sum_lw / ctrl->sum_w;
}

// ---------------------------------------------------------------------------
// Host launcher. Workspace layout: [pred_bits: PIX u32][nll: PIX f32][Ctrl]
// -> 16.8 MB. All kernels on `stream`; deterministic; re-inits all state.
// ---------------------------------------------------------------------------
extern "C" void kernel_launch(void* const* d_in, const int* in_sizes, int n_in,
                              void* d_out, int out_size, void* d_ws, size_t ws_size,
                              hipStream_t stream) {
    const float* predict = (const float*)d_in[0];
    const int*   target  = (const int*)d_in[1];
    const float* weight  = (const float*)d_in[2];

    unsigned int* pred_bits = (unsigned int*)d_ws;
    float*        nll       = (float*)((char*)d_ws + (size_t)PIX * 4);
    Ctrl*         ctrl      = (Ctrl*)((char*)d_ws + (size_t)PIX * 8);

    ohem_init_kernel<<<1, 256, 0, stream>>>(ctrl);
    ohem_pass1_kernel<<<2048, 256, 0, stream>>>(predict, target, pred_bits, nll, ctrl);
    for (int pass = 0; pass < 4; ++pass) {
        ohem_hist_kernel<<<1024, 256, 0, stream>>>(pred_bits, ctrl, pass);
        ohem_select_kernel<<<1, 256, 0, stream>>>(ctrl, pass);
    }
    ohem_final_kernel<<<1024, 256, 0, stream>>>(target, weight, pred_bits, nll, ctrl);
    ohem_finalize_kernel<<<1, 1, 0, stream>>>(ctrl, (float*)d_out);
}